// DenseCapsBayesian_20246475833891
// MI455X (gfx1250) — compile-verified
//
#include <hip/hip_runtime.h>

// ---------------------------------------------------------------------------
// DenseCapsBayesian on MI455X (gfx1250, wave32)
//   B=64, IN_CAPS=2048, IN_DIMS=8, N_CAPS=32, DIMS=16, R_ITER=3
// einsum via V_WMMA_F32_16X16X4_F32 (fp32 native matrix path),
// routing fused to 2 passes over u_hat (iter0 softmax is uniform 1/32).
// u_hat kept in WMMA-native layout [i][j][nt][lane][vr] so all u_hat traffic
// is 32B-per-lane vectorized (b128 stores/loads), both producer & consumer.
// ---------------------------------------------------------------------------

typedef __attribute__((ext_vector_type(2))) float v2f;
typedef __attribute__((ext_vector_type(8))) float v8f;

#define BATCH     64
#define IN_CAPS   2048
#define IN_DIMS   8
#define N_CAPS    32
#define DIMS      16
#define JD        512            // N_CAPS*DIMS
#define S_ELEMS   (BATCH * N_CAPS * DIMS)   // 32768
#define EPS_K     1e-7f
#define SOFTPLUS_C 0.5413248546129181f      // log(expm1(1))

#define IPB       8    // input caps per einsum workgroup  -> 256 WGs
#define ICHUNK    32   // input caps per routing workgroup -> 64 chunks

__device__ __forceinline__ v8f v8f_zero() {
  v8f z;
#pragma unroll
  for (int e = 0; e < 8; ++e) z[e] = 0.0f;
  return z;
}

// ---------------------------------------------------------------------------
// Kernel 1: W = loc + (1e-5 + softplus(C + scale_raw)) * eps
// ---------------------------------------------------------------------------
__global__ void k_weights(const float* __restrict__ loc,
                          const float* __restrict__ sraw,
                          const float* __restrict__ eps,
                          float* __restrict__ W, int n) {
  int idx = blockIdx.x * blockDim.x + threadIdx.x;
  if (idx >= n) return;
  float t  = SOFTPLUS_C + sraw[idx];
  float sp = (t > 20.0f) ? t : log1pf(__expf(t));
  W[idx] = loc[idx] + (1e-5f + sp) * eps[idx];
}

// ---------------------------------------------------------------------------
// Kernel 2: u_hat[b,i,j,d] = sum_k W[i,jd,k] * x[b,i,k]  via WMMA f32 16x16x4
//   Per input cap i: GEMM (M=512 jd) x (K=8) x (N=64 batch).
//   Wave w owns M-tiles mt = 4w..4w+3; loops all 4 N-tiles; K=8 -> 2 WMMAs.
//   Tile element (lane, vr): batch b = nt*16 + (lane&15),
//                            jd = mt*16 + (lane>>4)*8 + vr  (j = mt, d = rest)
//   Store per lane = 8 contiguous floats -> b128 traffic.
//   Also accumulates sum_i u_hat (iter0: c uniform) -> per-WG partial.
// ---------------------------------------------------------------------------
__global__ void __launch_bounds__(256)
k_uhat_wmma(const float* __restrict__ W, const float* __restrict__ x,
            float* __restrict__ uhat, float* __restrict__ spart0) {
  const int tid = threadIdx.x;
  const int w   = tid >> 5;
  const int l   = tid & 31;
  const int b16 = l & 15;
  const int hi  = l >> 4;
  const int i0  = blockIdx.x * IPB;

  v8f sum[4][4];
#pragma unroll
  for (int mtl = 0; mtl < 4; ++mtl)
#pragma unroll
    for (int nt = 0; nt < 4; ++nt) sum[mtl][nt] = v8f_zero();

  for (int ii = 0; ii < IPB; ++ii) {
    const int i = i0 + ii;

    v2f afrag[4][2];
#pragma unroll
    for (int mtl = 0; mtl < 4; ++mtl) {
      const int jd = (w * 4 + mtl) * 16 + b16;         // A row (M)
      const float* wp = W + ((size_t)i * JD + jd) * IN_DIMS;
#pragma unroll
      for (int kt = 0; kt < 2; ++kt)
        afrag[mtl][kt] = *(const v2f*)(wp + kt * 4 + 2 * hi);
    }

    v2f bfrag[4][2];
#pragma unroll
    for (int nt = 0; nt < 4; ++nt) {
      const int b = nt * 16 + b16;                     // B col (N)
      const float* xp = x + ((size_t)b * IN_CAPS + i) * IN_DIMS;
#pragma unroll
      for (int kt = 0; kt < 2; ++kt)
        bfrag[nt][kt] = *(const v2f*)(xp + kt * 4 + 2 * hi);
    }

#pragma unroll
    for (int mtl = 0; mtl < 4; ++mtl) {
#pragma unroll
      for (int nt = 0; nt < 4; ++nt) {
        v8f acc = v8f_zero();
        acc = __builtin_amdgcn_wmma_f32_16x16x4_f32(
            false, afrag[mtl][0], false, bfrag[nt][0], (short)0, acc, false, false);
        acc = __builtin_amdgcn_wmma_f32_16x16x4_f32(
            false, afrag[mtl][1], false, bfrag[nt][1], (short)0, acc, false, false);
        sum[mtl][nt] += acc;
        // [i][j][nt][lane][vr]: lane owns 8 contiguous floats (32B)
        float* up = uhat +
            (((((size_t)i * N_CAPS + (w * 4 + mtl)) * 4 + nt) * 32) + l) * 8;
        *(v8f*)up = acc;
      }
    }
  }

  // per-WG partial of sum_i u_hat, natural [b][j][d]; d-contiguous per lane
  float* sp = spart0 + (size_t)blockIdx.x * S_ELEMS;
#pragma unroll
  for (int mtl = 0; mtl < 4; ++mtl) {
    const int j = w * 4 + mtl;
#pragma unroll
    for (int nt = 0; nt < 4; ++nt) {
      const int b = nt * 16 + b16;
      *(v8f*)(sp + ((size_t)b * N_CAPS + j) * DIMS + hi * 8) = sum[mtl][nt];
    }
  }
}

// ---------------------------------------------------------------------------
// Kernel 3: deterministic reduce of partials + squash -> v
//   one thread per (b,j) row of 16 dims
// ---------------------------------------------------------------------------
__global__ void k_squash(const float* __restrict__ parts, int nparts, float scale,
                         float* __restrict__ vout) {
  const int row = blockIdx.x * blockDim.x + threadIdx.x;   // b*32+j
  if (row >= BATCH * N_CAPS) return;
  float s[DIMS];
#pragma unroll
  for (int d = 0; d < DIMS; ++d) s[d] = 0.0f;
  for (int p = 0; p < nparts; ++p) {
    const float* pp = parts + (size_t)p * S_ELEMS + (size_t)row * DIMS;
#pragma unroll
    for (int d = 0; d < DIMS; ++d) s[d] += pp[d];
  }
  float n2 = 0.0f;
#pragma unroll
  for (int d = 0; d < DIMS; ++d) { s[d] *= scale; n2 += s[d] * s[d]; }
  const float f = (n2 / (1.0f + n2)) / sqrtf(n2 + EPS_K);
#pragma unroll
  for (int d = 0; d < DIMS; ++d) vout[(size_t)row * DIMS + d] = s[d] * f;
}

// ---------------------------------------------------------------------------
// Kernel 4: fused routing pass (one per remaining iteration).
//   grid = (nt=4, chunk=64); WG owns 16 batches (one nt) and 32 input caps.
//   Wave w owns j = 4w..4w+3; lane&15 = batch, lane>>4 = d-half.
//   FINAL=false: b1 = u.v0, store logits, c1 -> s1 partials
//   FINAL=true : b2 = b1 + u.v1, c2 -> d_out (routing weights) + s2 partials
// ---------------------------------------------------------------------------
#define VSTRIDE 520   // 512 padded (520 % 64 != 0 -> no LDS bank conflicts)
#define TSTRIDE 33

template <bool FINAL>
__global__ void __launch_bounds__(256)
k_route(const float* __restrict__ uhat, const float* __restrict__ vin,
        float* __restrict__ blog, float* __restrict__ spart,
        float* __restrict__ c_out) {
  __shared__ float vsh[16 * VSTRIDE];
  __shared__ float tsh[16 * TSTRIDE];

  const int tid   = threadIdx.x;
  const int w     = tid >> 5;
  const int l     = tid & 31;
  const int b16   = l & 15;
  const int hi    = l >> 4;
  const int nt    = blockIdx.x;
  const int chunk = blockIdx.y;
  const int i0    = chunk * ICHUNK;

  for (int idx = tid; idx < 16 * JD; idx += 256)
    vsh[(idx >> 9) * VSTRIDE + (idx & 511)] =
        vin[(size_t)nt * 16 * JD + idx];
  __syncthreads();

  v8f sacc[4];
#pragma unroll
  for (int jl = 0; jl < 4; ++jl) sacc[jl] = v8f_zero();

  for (int ii = 0; ii < ICHUNK; ++ii) {
    const int i = i0 + ii;
    v8f u[4];
    float tpart[4];
#pragma unroll
    for (int jl = 0; jl < 4; ++jl) {
      const int j = w * 4 + jl;
      const float* up =
          uhat + (((((size_t)i * N_CAPS + j) * 4 + nt) * 32) + l) * 8;
      const v8f uv = *(const v8f*)up;               // b128 pair, coalesced
      u[jl] = uv;
      float tp = 0.0f;
#pragma unroll
      for (int vr = 0; vr < 8; ++vr)
        tp += uv[vr] * vsh[b16 * VSTRIDE + j * DIMS + hi * 8 + vr];
      tpart[jl] = tp;
    }
#pragma unroll
    for (int jl = 0; jl < 4; ++jl) {
      // combine the two d-halves (lane ^ 16) -> full 16-dim dot
      const float t = tpart[jl] + __shfl_xor(tpart[jl], 16, 32);
      const int j = w * 4 + jl;
      const size_t boff = (size_t)nt * (IN_CAPS * N_CAPS * 16) +
                          ((size_t)i * N_CAPS + j) * 16 + b16;
      float bl;
      if (FINAL) {
        bl = blog[boff] + t;
      } else {
        bl = t;
        if (hi == 0) blog[boff] = t;
      }
      if (hi == 0) tsh[b16 * TSTRIDE + j] = bl;
    }
    __syncthreads();

    float m = -3.4e38f;
#pragma unroll
    for (int j = 0; j < N_CAPS; ++j) m = fmaxf(m, tsh[b16 * TSTRIDE + j]);
    float den = 0.0f;
#pragma unroll
    for (int j = 0; j < N_CAPS; ++j) den += __expf(tsh[b16 * TSTRIDE + j] - m);
    const float rden = 1.0f / den;

#pragma unroll
    for (int jl = 0; jl < 4; ++jl) {
      const int j = w * 4 + jl;
      const float c = __expf(tsh[b16 * TSTRIDE + j] - m) * rden;
      if (FINAL && hi == 0) {
        const int b = nt * 16 + b16;
        c_out[((size_t)b * IN_CAPS + i) * N_CAPS + j] = c;
      }
      sacc[jl] += u[jl] * c;
    }
    __syncthreads();
  }

  float* sp = spart + (size_t)chunk * S_ELEMS;
  const int b = nt * 16 + b16;
#pragma unroll
  for (int jl = 0; jl < 4; ++jl) {
    const int j = w * 4 + jl;
    *(v8f*)(sp + ((size_t)b * N_CAPS + j) * DIMS + hi * 8) = sacc[jl];
  }
}

// ---------------------------------------------------------------------------
// Host-side orchestration
// ---------------------------------------------------------------------------
extern "C" void kernel_launch(void* const* d_in, const int* in_sizes, int n_in,
                              void* d_out, int out_size, void* d_ws, size_t ws_size,
                              hipStream_t stream) {
  const float* x    = (const float*)d_in[0];
  const float* loc  = (const float*)d_in[1];
  const float* sraw = (const float*)d_in[2];
  const float* eps  = (const float*)d_in[3];
  float* out = (float*)d_out;

  // workspace layout (floats)
  const size_t nW     = (size_t)IN_CAPS * N_CAPS * DIMS * IN_DIMS;   // 8,388,608
  const size_t nU     = (size_t)BATCH * IN_CAPS * N_CAPS * DIMS;     // 67,108,864
  const size_t nBlog  = (size_t)BATCH * IN_CAPS * N_CAPS;            // 4,194,304
  const size_t nSp0   = (size_t)(IN_CAPS / IPB) * S_ELEMS;           // 256 partials
  const size_t nSpR   = (size_t)(IN_CAPS / ICHUNK) * S_ELEMS;        // 64 partials

  float* ws      = (float*)d_ws;
  float* Wbuf    = ws;
  float* uhat    = Wbuf + nW;
  float* blog    = uhat + nU;
  float* spart0  = blog + nBlog;
  float* spartR  = spart0 + nSp0;
  float* v0      = spartR + nSpR;
  float* v1      = v0 + S_ELEMS;

  // 1) Bayesian weight sample
  k_weights<<<(int)((nW + 255) / 256), 256, 0, stream>>>(loc, sraw, eps, Wbuf, (int)nW);

  // 2) u_hat via WMMA (+ iter-0 partial sums, c uniform = 1/32)
  k_uhat_wmma<<<IN_CAPS / IPB, 256, 0, stream>>>(Wbuf, x, uhat, spart0);

  // 3) v0 = squash( (1/32) * sum_i u_hat )
  k_squash<<<(BATCH * N_CAPS + 255) / 256, 256, 0, stream>>>(
      spart0, IN_CAPS / IPB, 1.0f / (float)N_CAPS, v0);

  // 4) routing iteration 1: b1 = u.v0 ; c1 = softmax(b1) ; s1 partials
  k_route<false><<<dim3(4, IN_CAPS / ICHUNK), 256, 0, stream>>>(
      uhat, v0, blog, spartR, nullptr);

  // 5) v1 = squash(s1)
  k_squash<<<(BATCH * N_CAPS + 255) / 256, 256, 0, stream>>>(
      spartR, IN_CAPS / ICHUNK, 1.0f, v1);

  // 6) routing iteration 2: b2 = b1 + u.v1 ; c2 -> d_out ; s2 partials
  k_route<true><<<dim3(4, IN_CAPS / ICHUNK), 256, 0, stream>>>(
      uhat, v1, blog, spartR, out + S_ELEMS);

  // 7) v = squash(s2) -> d_out[0 : 32768]
  k_squash<<<(BATCH * N_CAPS + 255) / 256, 256, 0, stream>>>(
      spartR, IN_CAPS / ICHUNK, 1.0f, out);
}